// RobustGlobalPool2d_40767829574195
// MI455X (gfx1250) — compile-verified
//
#include <hip/hip_runtime.h>
#include <stdint.h>

// RobustGlobalPool2d: per-slice Newton solve of argmin_y sum phi(y - x),
// phi = pseudo-Huber (alpha = 1).  x: [8192 slices][4096 elems] fp32.
//
// MI455X design: one workgroup per slice; slice lives in registers
// (16 f32 per thread), HBM read exactly once (~134MB -> ~6us at 23.3TB/s).
// Staging uses the CDNA5 async global->LDS DMA path (ASYNCcnt), then 50
// (early-exiting) Newton iterations run on-chip: ~5 VALU + 1 v_rsq_f32 per
// element-iteration, wave32 butterfly + double-buffered LDS reduction
// (one workgroup barrier per iteration).

#define HW           4096
#define BLOCK        256
#define PER_THREAD   (HW / BLOCK)     // 16
#define NWAVES       (BLOCK / 32)     // 8 (wave32)
#define NEWTON_ITERS 50

#if __has_builtin(__builtin_amdgcn_global_load_async_to_lds_b128)
#define USE_ASYNC_LDS 1
#else
#define USE_ASYNC_LDS 0
#endif

typedef int v4i __attribute__((ext_vector_type(4)));
typedef __attribute__((address_space(1))) v4i* g_v4i_ptr;  // global
typedef __attribute__((address_space(3))) v4i* l_v4i_ptr;  // LDS

__device__ __forceinline__ float wave_reduce_add_f32(float v) {
    // wave32 butterfly
    v += __shfl_xor(v, 16, 32);
    v += __shfl_xor(v,  8, 32);
    v += __shfl_xor(v,  4, 32);
    v += __shfl_xor(v,  2, 32);
    v += __shfl_xor(v,  1, 32);
    return v;
}

__device__ __forceinline__ float fast_rsqrt(float t) {
#if __has_builtin(__builtin_amdgcn_rsqf)
    return __builtin_amdgcn_rsqf(t);   // v_rsq_f32
#else
    return rsqrtf(t);
#endif
}

__global__ __launch_bounds__(BLOCK)
void RobustGlobalPool2d_newton_kernel(const float* __restrict__ x,
                                      float* __restrict__ out) {
    const int tid  = threadIdx.x;
    const int wave = tid >> 5;
    const int lane = tid & 31;
    const size_t slice = blockIdx.x;
    const float* __restrict__ xs = x + slice * (size_t)HW;

    // Double-buffered cross-wave partials: reads of buffer b at step i are
    // separated from the next writes of buffer b (step i+2) by step i+1's
    // barrier, so no trailing WAR barrier is needed.
    __shared__ float red_g[2][NWAVES];
    __shared__ float red_h[2][NWAVES];

    float v[PER_THREAD];

#if USE_ASYNC_LDS
    __shared__ float tile[HW];
    {
        // Each lane DMAs 16B per issue; 4 issues * 256 lanes * 16B = 16KB.
        char* lgen = (char*)&tile[0];
        #pragma unroll
        for (int p = 0; p < 4; ++p) {
            const int byteOff = (p * BLOCK + tid) * 16;
            g_v4i_ptr gsrc = (g_v4i_ptr)(uintptr_t)((const char*)xs + byteOff);
            l_v4i_ptr ldst = (l_v4i_ptr)(uint32_t)(uintptr_t)(lgen + byteOff);
            __builtin_amdgcn_global_load_async_to_lds_b128(gsrc, ldst, 0, 0);
        }
#if __has_builtin(__builtin_amdgcn_s_wait_asynccnt)
        __builtin_amdgcn_s_wait_asynccnt(0);
#else
        asm volatile("s_wait_asynccnt 0x0" ::: "memory");
#endif
    }
    __syncthreads();
    // Strided element ownership: conflict-free ds_load_b32 (stride 4B/lane).
    #pragma unroll
    for (int j = 0; j < PER_THREAD; ++j) v[j] = tile[j * BLOCK + tid];
#else
    // Direct coalesced global loads (b32, 128B per wave-instruction).
    #pragma unroll
    for (int j = 0; j < PER_THREAD; ++j) v[j] = xs[j * BLOCK + tid];
#endif

    // ---- init: y0 = mean over the slice (uses buffer 1, i.e. step -1) ----
    float s = 0.0f;
    #pragma unroll
    for (int j = 0; j < PER_THREAD; ++j) s += v[j];
    s = wave_reduce_add_f32(s);
    if (lane == 0) red_g[1][wave] = s;
    __syncthreads();
    float tot = 0.0f;
    #pragma unroll
    for (int w = 0; w < NWAVES; ++w) tot += red_g[1][w];
    float y = tot * (1.0f / (float)HW);

    // ---- Newton iterations (uniform early exit at the fixed point) ----
    for (int it = 0; it < NEWTON_ITERS; ++it) {
        const int buf = it & 1;
        float g = 0.0f, h = 0.0f;
        #pragma unroll
        for (int j = 0; j < PER_THREAD; ++j) {
            float z  = y - v[j];
            float t  = __builtin_fmaf(z, z, 1.0f);   // alpha = 1
            float r  = fast_rsqrt(t);                // 1/sqrt(t)
            g = __builtin_fmaf(z, r, g);             // z / sqrt(t)
            float r2 = r * r;
            h = __builtin_fmaf(r2, r, h);            // t^-1.5
        }
        g = wave_reduce_add_f32(g);
        h = wave_reduce_add_f32(h);
        if (lane == 0) { red_g[buf][wave] = g; red_h[buf][wave] = h; }
        __syncthreads();
        float G = 0.0f, H = 0.0f;
        #pragma unroll
        for (int w = 0; w < NWAVES; ++w) { G += red_g[buf][w]; H += red_h[buf][w]; }
        float y_new = y - G / fmaxf(H, 1e-12f);  // identical on every thread
        if (y_new == y) break;   // fixed point: remaining steps are identity
        y = y_new;
    }

    if (tid == 0) out[slice] = y;
}

extern "C" void kernel_launch(void* const* d_in, const int* in_sizes, int n_in,
                              void* d_out, int out_size, void* d_ws, size_t ws_size,
                              hipStream_t stream) {
    const float* x = (const float*)d_in[0];
    float* out = (float*)d_out;
    const int n_slices = out_size;   // 8192 = 32*256; HW fixed at 4096
    (void)in_sizes; (void)n_in; (void)d_ws; (void)ws_size;
    RobustGlobalPool2d_newton_kernel<<<dim3(n_slices), dim3(BLOCK), 0, stream>>>(x, out);
}